// Stabilized_Entropic_Wasserstein_26706106647245
// MI455X (gfx1250) — compile-verified
//
#include <hip/hip_runtime.h>

typedef float v2f __attribute__((ext_vector_type(2)));
typedef float v8f __attribute__((ext_vector_type(8)));

constexpr int   N       = 4096;
constexpr int   M       = 4096;
constexpr int   D       = 64;
constexpr float EPS     = 0.1f;
constexpr float INV_EPS = 10.0f;

// ---------------------------------------------------------------------------
// Kernel 0: row norms of x and y, plus zero-init of g (f is written before read)
// ---------------------------------------------------------------------------
__global__ __launch_bounds__(256)
void norms_init_kernel(const float* __restrict__ x, const float* __restrict__ y,
                       float* __restrict__ xn, float* __restrict__ yn,
                       float* __restrict__ g) {
    int t = blockIdx.x * 256 + threadIdx.x;   // 0 .. N+M-1
    if (t < N) {
        const float* r = x + (size_t)t * D;
        float s = 0.f;
        #pragma unroll
        for (int k = 0; k < D; ++k) s += r[k] * r[k];
        xn[t] = s;
        g[t]  = 0.f;
    } else {
        int j = t - N;
        const float* r = y + (size_t)j * D;
        float s = 0.f;
        #pragma unroll
        for (int k = 0; k < D; ++k) s += r[k] * r[k];
        yn[j] = s;
    }
}

// ---------------------------------------------------------------------------
// Kernel 1: C = xn_i + yn_j - 2 * (x . y^T) via V_WMMA_F32_16X16X4_F32
// One wave (32 threads) computes one 16x16 tile; K=64 -> 16 wmma ops.
// ---------------------------------------------------------------------------
__global__ __launch_bounds__(32)
void cost_wmma_kernel(const float* __restrict__ x, const float* __restrict__ y,
                      const float* __restrict__ xn, const float* __restrict__ yn,
                      float* __restrict__ C) {
    const int lane  = threadIdx.x;            // 0..31, EXEC all ones
    const int laneM = lane & 15;
    const int kb    = (lane >> 4) << 1;       // 0 or 2 (K sub-offset per half-wave)
    const int row0  = blockIdx.y * 16;
    const int col0  = blockIdx.x * 16;

    const float* xr = x + (size_t)(row0 + laneM) * D + kb;
    const float* yr = y + (size_t)(col0 + laneM) * D + kb;

    v8f acc = {};
    #pragma unroll
    for (int k0 = 0; k0 < D; k0 += 4) {
        v2f a = *(const v2f*)(xr + k0);       // A: 16x4 fp32 tile slice
        v2f b = *(const v2f*)(yr + k0);       // B: 4x16 fp32 tile slice (y^T)
        acc = __builtin_amdgcn_wmma_f32_16x16x4_f32(
            /*neg_a=*/false, a, /*neg_b=*/false, b,
            /*c_mod=*/(short)0, acc, /*reuse_a=*/false, /*reuse_b=*/false);
    }

    // D layout: VGPR r holds row (row0 + r + 8*(lane>=16)), col = lane&15
    const int n   = laneM;
    const int mhi = (lane >> 4) << 3;
    const float yv = yn[col0 + n];
    #pragma unroll
    for (int r = 0; r < 8; ++r) {
        int i = row0 + mhi + r;
        C[(size_t)i * M + col0 + n] = xn[i] + yv - 2.0f * acc[r];
    }
}

// ---------------------------------------------------------------------------
// Online (max,sum) logsumexp helpers
// ---------------------------------------------------------------------------
__device__ __forceinline__ void lse_push(float v, float& m, float& s) {
    if (v > m) { s = s * __expf(m - v) + 1.0f; m = v; }
    else       { s += __expf(v - m); }
}
__device__ __forceinline__ void lse_merge(float& m1, float& s1, float m2, float s2) {
    if (m1 >= m2) { s1 = s1 + s2 * __expf(m2 - m1); }
    else          { s1 = s2 + s1 * __expf(m1 - m2); m1 = m2; }
}

// ---------------------------------------------------------------------------
// Kernel 2: f_i = EPS * (log p_i - LSE_j((g_j - C_ij)/EPS)).  One block per row.
// ---------------------------------------------------------------------------
__global__ __launch_bounds__(256)
void row_lse_kernel(const float* __restrict__ C, const float* __restrict__ g,
                    const float* __restrict__ p, float* __restrict__ f) {
    __shared__ float sm[256], ss[256];
    const int i   = blockIdx.x;
    const int tid = threadIdx.x;
    const float* row = C + (size_t)i * M;

    float m = -INFINITY, s = 0.f;
    for (int j = tid; j < M; j += 256)
        lse_push((g[j] - row[j]) * INV_EPS, m, s);

    sm[tid] = m; ss[tid] = s;
    __syncthreads();
    for (int st = 128; st >= 1; st >>= 1) {
        if (tid < st) {
            float m1 = sm[tid], s1 = ss[tid];
            lse_merge(m1, s1, sm[tid + st], ss[tid + st]);
            sm[tid] = m1; ss[tid] = s1;
        }
        __syncthreads();
    }
    if (tid == 0)
        f[i] = EPS * (__logf(p[i]) - (sm[0] + __logf(ss[0])));
}

// ---------------------------------------------------------------------------
// Kernel 3: g_j = EPS * (log q_j - LSE_i((f_i - C_ij)/EPS)).
// Block = 32 columns x 8 row-groups -> coalesced column reduction, no transpose.
// ---------------------------------------------------------------------------
__global__ __launch_bounds__(256)
void col_lse_kernel(const float* __restrict__ C, const float* __restrict__ fv,
                    const float* __restrict__ q, float* __restrict__ g) {
    __shared__ float sm[256], ss[256];
    const int tid = threadIdx.x;
    const int jl  = tid & 31;
    const int ig  = tid >> 5;
    const int j   = blockIdx.x * 32 + jl;

    float m = -INFINITY, s = 0.f;
    for (int i = ig; i < N; i += 8)
        lse_push((fv[i] - C[(size_t)i * M + j]) * INV_EPS, m, s);

    sm[tid] = m; ss[tid] = s;
    __syncthreads();
    for (int st = 4; st >= 1; st >>= 1) {
        if (ig < st) {
            float m1 = sm[tid], s1 = ss[tid];
            lse_merge(m1, s1, sm[tid + st * 32], ss[tid + st * 32]);
            sm[tid] = m1; ss[tid] = s1;
        }
        __syncthreads();
    }
    if (ig == 0)
        g[j] = EPS * (__logf(q[j]) - (sm[0 * 32 + jl] + __logf(ss[0 * 32 + jl])));
}

// ---------------------------------------------------------------------------
// Kernel 4: P = exp((f_i + g_j - C)/EPS) in-place over C; per-block partial of
// sum(P*C).  4 elements/thread, 1024 elements/block.
// ---------------------------------------------------------------------------
__global__ __launch_bounds__(256)
void finalize_kernel(float* __restrict__ C, const float* __restrict__ f,
                     const float* __restrict__ g, float* __restrict__ partials) {
    __shared__ float sd[256];
    const int tid = threadIdx.x;
    size_t base = (size_t)blockIdx.x * 1024 + tid;

    float acc = 0.f;
    #pragma unroll
    for (int t = 0; t < 4; ++t) {
        size_t e = base + (size_t)t * 256;
        int i = (int)(e >> 12);        // e / 4096
        int j = (int)(e & 4095);       // e % 4096
        float c  = C[e];
        float pv = __expf((f[i] + g[j] - c) * INV_EPS);
        C[e] = pv;                     // overwrite C with P in-place
        acc += pv * c;
    }

    sd[tid] = acc;
    __syncthreads();
    for (int st = 128; st >= 1; st >>= 1) {
        if (tid < st) sd[tid] += sd[tid + st];
        __syncthreads();
    }
    if (tid == 0) partials[blockIdx.x] = sd[0];
}

// ---------------------------------------------------------------------------
// Kernel 5: deterministic final sum of 16384 partials -> d_out[0]
// ---------------------------------------------------------------------------
__global__ __launch_bounds__(256)
void sum_partials_kernel(const float* __restrict__ partials, float* __restrict__ out,
                         int nPartials) {
    __shared__ float sd[256];
    const int tid = threadIdx.x;
    float a = 0.f;
    for (int k = tid; k < nPartials; k += 256) a += partials[k];
    sd[tid] = a;
    __syncthreads();
    for (int st = 128; st >= 1; st >>= 1) {
        if (tid < st) sd[tid] += sd[tid + st];
        __syncthreads();
    }
    if (tid == 0) out[0] = sd[0];
}

// ---------------------------------------------------------------------------
extern "C" void kernel_launch(void* const* d_in, const int* in_sizes, int n_in,
                              void* d_out, int out_size, void* d_ws, size_t ws_size,
                              hipStream_t stream) {
    const float* x = (const float*)d_in[0];   // [N, D]
    const float* y = (const float*)d_in[1];   // [M, D]
    const float* p = (const float*)d_in[2];   // [N]
    const float* q = (const float*)d_in[3];   // [M]

    float* out  = (float*)d_out;              // [0] = w_cost, [1..] = P (NxM)
    float* Cmat = out + 1;                    // C materialized where P will live

    float* wsf      = (float*)d_ws;
    float* xn       = wsf;                    // [N]
    float* yn       = wsf + N;                // [M]
    float* f        = wsf + N + M;            // [N]
    float* g        = wsf + 2 * N + M;        // [M]
    float* partials = wsf + 2 * N + 2 * M;    // [16384]

    // 0: norms + g = 0
    norms_init_kernel<<<(N + M) / 256, 256, 0, stream>>>(x, y, xn, yn, g);

    // 1: C via fp32 WMMA (one wave per 16x16 tile)
    dim3 gemmGrid(M / 16, N / 16);
    cost_wmma_kernel<<<gemmGrid, 32, 0, stream>>>(x, y, xn, yn, Cmat);

    // 2/3: 10 Sinkhorn iterations (C stays resident in the 192MB L2)
    for (int it = 0; it < 10; ++it) {
        row_lse_kernel<<<N, 256, 0, stream>>>(Cmat, g, p, f);
        col_lse_kernel<<<M / 32, 256, 0, stream>>>(Cmat, f, q, g);
    }

    // 4/5: P = exp(S) in-place, w_cost = sum(P*C) via deterministic 2-stage sum
    const int nBlocks = (N * M) / 1024;       // 16384
    finalize_kernel<<<nBlocks, 256, 0, stream>>>(Cmat, f, g, partials);
    sum_partials_kernel<<<1, 256, 0, stream>>>(partials, out, nBlocks);
}